// GraphEncoder_44255343018143
// MI455X (gfx1250) — compile-verified
//
#include <hip/hip_runtime.h>
#include <hip/hip_bf16.h>

// ---------------------------------------------------------------------------
// HGT forward for MI455X (gfx1250, wave32).  fp32 WMMA (V_WMMA_F32_16X16X4_F32)
// for all dense math; relation transforms hoisted from per-edge to per-node;
// segment softmax via encoded-u32 atomicMax + f32 atomicAdd; wave-per-(edge,head)
// scatter aggregation.
// ---------------------------------------------------------------------------

#define NGENE 50000
#define NTRAIT 20000
#define HID 128

typedef __attribute__((ext_vector_type(2))) float v2f;
typedef __attribute__((ext_vector_type(8))) float v8f;

// monotone float<->uint order-preserving encoding (memset-0 init safe: any
// finite score encodes > 0, and m[] is only read for dsts with >=1 edge)
__device__ __forceinline__ unsigned encF(float x) {
  unsigned v = __float_as_uint(x);
  return (v & 0x80000000u) ? ~v : (v | 0x80000000u);
}
__device__ __forceinline__ float decF(unsigned e) {
  return (e & 0x80000000u) ? __uint_as_float(e & 0x7fffffffu)
                           : __uint_as_float(~e);
}

// ---------------------------------------------------------------------------
// Generic WMMA GEMM:  C[M,N] = A[M,K] @ W[K,N] + bias   (N == 128 always here)
// MODE 0: bias only.  MODE 1: h_out = relu(a*(acc+bias) + (2-a)*h_prev),
//         a = sigmoid(*skip)   (fused HGT skip-blend + residual + relu)
// block = (32,4): 4 waves, wave ty owns a 16x16 output tile; grid=(M/16, N/64).
// M is always a multiple of 16, K a multiple of 4 -> EXEC stays all-ones (WMMA req).
// ---------------------------------------------------------------------------
template <int MODE>
__global__ __launch_bounds__(128) void wmma_gemm(
    const float* __restrict__ A, const float* __restrict__ W,
    const float* __restrict__ bias, float* __restrict__ C, int M, int K, int N,
    const float* __restrict__ hprev, const float* __restrict__ skip) {
  const int lane = threadIdx.x;                       // 0..31
  const int m0 = blockIdx.x * 16;
  const int n0 = (blockIdx.y * 4 + threadIdx.y) * 16;
  const int r = lane & 15;
  const int ksel = (lane >> 4) * 2;                   // lanes 0-15 -> K+0,1 ; 16-31 -> K+2,3
  const float* Arow = A + (size_t)(m0 + r) * K;
  const float* Wcol = W + n0 + r;
  v8f acc = {0.f, 0.f, 0.f, 0.f, 0.f, 0.f, 0.f, 0.f};
  for (int k0 = 0; k0 < K; k0 += 4) {
    const int kk = k0 + ksel;
    v2f a, b;
    a.x = Arow[kk];
    a.y = Arow[kk + 1];
    b.x = Wcol[(size_t)kk * N];
    b.y = Wcol[(size_t)(kk + 1) * N];
    acc = __builtin_amdgcn_wmma_f32_16x16x4_f32(false, a, false, b, (short)0,
                                                acc, false, false);
  }
  const float bval = bias ? bias[n0 + r] : 0.f;
  float aS = 0.f, bS = 0.f;
  if (MODE == 1) {
    const float sv = *skip;
    aS = 1.f / (1.f + __expf(-sv));
    bS = 2.f - aS;
  }
  const int roff = (lane >> 4) * 8;  // C layout: vgpr j -> M=j (lanes<16) / j+8
#pragma unroll
  for (int j = 0; j < 8; ++j) {
    const size_t idx = (size_t)(m0 + j + roff) * N + n0 + r;
    float o = acc[j] + bval;
    if (MODE == 1) {
      o = aS * o + bS * hprev[idx];
      o = o > 0.f ? o : 0.f;
    }
    C[idx] = o;
  }
}

// ---------------------------------------------------------------------------
// Per-head relation transform: Y[n, h*32+f] = sum_d X[n, h*32+d] * R[h,d,f]
// (hoisted einsum 'ehd,hdf->ehf').  block=(32,8): wave -> (head h, col-half).
// ---------------------------------------------------------------------------
__global__ __launch_bounds__(256) void wmma_rel(const float* __restrict__ X,
                                                const float* __restrict__ R,
                                                float* __restrict__ Y, int M) {
  const int lane = threadIdx.x;
  const int wv = threadIdx.y;  // 0..7
  const int h = wv >> 1;
  const int nt = wv & 1;
  const int m0 = blockIdx.x * 16;
  const int r = lane & 15;
  const int ksel = (lane >> 4) * 2;
  const float* Xrow = X + (size_t)(m0 + r) * HID + h * 32;
  const float* Rh = R + h * 1024 + nt * 16 + r;
  v8f acc = {0.f, 0.f, 0.f, 0.f, 0.f, 0.f, 0.f, 0.f};
#pragma unroll
  for (int k0 = 0; k0 < 32; k0 += 4) {
    const int kk = k0 + ksel;
    v2f a, b;
    a.x = Xrow[kk];
    a.y = Xrow[kk + 1];
    b.x = Rh[kk * 32];
    b.y = Rh[(kk + 1) * 32];
    acc = __builtin_amdgcn_wmma_f32_16x16x4_f32(false, a, false, b, (short)0,
                                                acc, false, false);
  }
  const int roff = (lane >> 4) * 8;
#pragma unroll
  for (int j = 0; j < 8; ++j)
    Y[(size_t)(m0 + j + roff) * HID + h * 32 + nt * 16 + r] = acc[j];
}

// ---------------------------------------------------------------------------
// Row LayerNorm over 128 features; one wave per row (wave32: 4 elems / lane).
// ---------------------------------------------------------------------------
__global__ __launch_bounds__(256) void layernorm128(float* __restrict__ X,
                                                    const float* __restrict__ g,
                                                    const float* __restrict__ b,
                                                    int M) {
  const int row = blockIdx.x * 8 + threadIdx.y;
  if (row >= M) return;
  const int lane = threadIdx.x;
  float* xr = X + (size_t)row * HID;
  float v0 = xr[lane], v1 = xr[lane + 32], v2 = xr[lane + 64],
        v3 = xr[lane + 96];
  float s = v0 + v1 + v2 + v3;
  for (int off = 16; off > 0; off >>= 1) s += __shfl_xor(s, off, 32);
  const float mean = s * (1.f / 128.f);
  const float d0 = v0 - mean, d1 = v1 - mean, d2 = v2 - mean, d3 = v3 - mean;
  float q = d0 * d0 + d1 * d1 + d2 * d2 + d3 * d3;
  for (int off = 16; off > 0; off >>= 1) q += __shfl_xor(q, off, 32);
  const float inv = rsqrtf(q * (1.f / 128.f) + 1e-5f);
  xr[lane] = d0 * inv * g[lane] + b[lane];
  xr[lane + 32] = d1 * inv * g[lane + 32] + b[lane + 32];
  xr[lane + 64] = d2 * inv * g[lane + 64] + b[lane + 64];
  xr[lane + 96] = d3 * inv * g[lane + 96] + b[lane + 96];
}

// ---------------------------------------------------------------------------
// Edge scores: s[e,h] = (q[dst,h,:] . krel[src,h,:]) * p_rel[h] / sqrt(32)
// + running segment max via encoded-u32 atomicMax.
// ---------------------------------------------------------------------------
__global__ __launch_bounds__(256) void score_k(
    const int* __restrict__ src, const int* __restrict__ dst,
    const float* __restrict__ Q, const float* __restrict__ KR,
    const float* __restrict__ prel, float* __restrict__ S,
    unsigned* __restrict__ mEnc, int E) {
  const int t = blockIdx.x * 256 + threadIdx.x;
  if (t >= E * 4) return;
  const int e = t >> 2, h = t & 3;
  const int sI = src[e], dI = dst[e];
  const float4* qr = (const float4*)(Q + (size_t)dI * HID + h * 32);
  const float4* kr = (const float4*)(KR + (size_t)sI * HID + h * 32);
  __builtin_prefetch(kr, 0, 0);  // global_prefetch_b8
  float acc = 0.f;
#pragma unroll
  for (int i = 0; i < 8; ++i) {
    const float4 a = qr[i], b = kr[i];
    acc += a.x * b.x + a.y * b.y + a.z * b.z + a.w * b.w;
  }
  const float sv = acc * prel[h] * 0.17677669529663687f;  // 1/sqrt(32)
  S[t] = sv;
  atomicMax(mEnc + (size_t)dI * 4 + h, encF(sv));
}

// e = exp(s - m[dst]);  S <- e;  den[dst,h] += e
__global__ __launch_bounds__(256) void expsum_k(const int* __restrict__ dst,
                                                float* __restrict__ S,
                                                const unsigned* __restrict__ mEnc,
                                                float* __restrict__ den, int E) {
  const int t = blockIdx.x * 256 + threadIdx.x;
  if (t >= E * 4) return;
  const int e = t >> 2, h = t & 3;
  const int dI = dst[e];
  const float m = decF(mEnc[(size_t)dI * 4 + h]);
  const float ev = __expf(S[t] - m);
  S[t] = ev;
  atomicAdd(den + (size_t)dI * 4 + h, ev);
}

// One wave per (edge, head): 32 lanes scatter-add the 32-float head slice.
__global__ __launch_bounds__(256) void agg_k(
    const int* __restrict__ src, const int* __restrict__ dst,
    const float* __restrict__ S, const float* __restrict__ den,
    const float* __restrict__ VR, float* __restrict__ AG, int E) {
  const int gw = blockIdx.x * 8 + (threadIdx.x >> 5);
  if (gw >= E * 4) return;
  const int lane = threadIdx.x & 31;
  const int e = gw >> 2, h = gw & 3;
  const int sI = src[e], dI = dst[e];
  const float* vp = VR + (size_t)sI * HID + h * 32;
  __builtin_prefetch(vp + lane, 0, 0);
  const float alpha = S[gw] / (den[(size_t)dI * 4 + h] + 1e-16f);
  atomicAdd(AG + (size_t)dI * HID + h * 32 + lane, alpha * vp[lane]);
}

// exact GELU (erf form), in place
__global__ __launch_bounds__(256) void gelu_k(float* __restrict__ X, int n) {
  const int t = blockIdx.x * 256 + threadIdx.x;
  if (t >= n) return;
  const float x = X[t];
  X[t] = 0.5f * x * (1.f + erff(x * 0.70710678118654752f));
}

// ---------------------------------------------------------------------------
// Host side
// ---------------------------------------------------------------------------
struct ConvIdx {
  int kgw, kgb, ktw, ktb, qgw, qgb, qtw, qtb, vgw, vgb, vtw, vtb;
  int ogw, ogb, otw, otb, skg, skt;
  int agt, atg, agg_, mgt, mtg, mgg, pgt, ptg, pgg;
};
struct Map {
  int xg, xt, wig, big, wit, bit, lngg, lngb, lntg, lntb;
  int owg, obg, owt, obt;
  int egts, egtd, etgs, etgd, eggs, eggd;
  ConvIdx c[2];
};

static void fillSortedConv(ConvIdx& c, int b) {
  c.agg_ = b + 0;  c.agt = b + 1;  c.atg = b + 2;
  c.kgw = b + 3;   c.kgb = b + 4;  c.ktw = b + 5;  c.ktb = b + 6;
  c.mgg = b + 7;   c.mgt = b + 8;  c.mtg = b + 9;
  c.ogw = b + 10;  c.ogb = b + 11; c.otw = b + 12; c.otb = b + 13;
  c.pgg = b + 14;  c.pgt = b + 15; c.ptg = b + 16;
  c.qgw = b + 17;  c.qgb = b + 18; c.qtw = b + 19; c.qtb = b + 20;
  c.skg = b + 21;  c.skt = b + 22;
  c.vgw = b + 23;  c.vgb = b + 24; c.vtw = b + 25; c.vtb = b + 26;
}
static void fillInsertConv(ConvIdx& c, int b) {
  c.kgw = b + 0;  c.kgb = b + 1;  c.ktw = b + 2;  c.ktb = b + 3;
  c.qgw = b + 4;  c.qgb = b + 5;  c.qtw = b + 6;  c.qtb = b + 7;
  c.vgw = b + 8;  c.vgb = b + 9;  c.vtw = b + 10; c.vtb = b + 11;
  c.ogw = b + 12; c.ogb = b + 13; c.otw = b + 14; c.otb = b + 15;
  c.skg = b + 16; c.skt = b + 17;
  c.agt = b + 18; c.atg = b + 19; c.agg_ = b + 20;
  c.mgt = b + 21; c.mtg = b + 22; c.mgg = b + 23;
  c.pgt = b + 24; c.ptg = b + 25; c.pgg = b + 26;
}

static Map resolveMap(const int* sz) {
  Map m{};
  if (sz[0] == 500000) {  // fully key-sorted pytree flatten
    m.eggd = 0; m.eggs = 1; m.egtd = 2; m.egts = 3; m.etgd = 4; m.etgs = 5;
    fillSortedConv(m.c[0], 6);
    fillSortedConv(m.c[1], 33);
    m.lngg = 60; m.lngb = 61; m.lntg = 62; m.lntb = 63;
    m.wig = 64;  m.big = 65;  m.wit = 66;  m.bit = 67;
    m.owg = 68;  m.obg = 69;  m.owt = 70;  m.obt = 71;
    m.xg = 72;   m.xt = 73;
  } else if (sz[2] == 65536) {  // full insertion order
    m.xg = 0; m.xt = 1;
    m.wig = 2; m.big = 3; m.wit = 4; m.bit = 5;
    m.lngg = 6; m.lngb = 7; m.lntg = 8; m.lntb = 9;
    fillInsertConv(m.c[0], 10);
    fillInsertConv(m.c[1], 37);
    m.owg = 64; m.obg = 65; m.owt = 66; m.obt = 67;
    m.egts = 68; m.egtd = 69; m.etgs = 70; m.etgd = 71; m.eggs = 72; m.eggd = 73;
  } else {  // top-level insertion, nested key-sorted
    m.xg = 0; m.xt = 1;
    fillSortedConv(m.c[0], 2);
    fillSortedConv(m.c[1], 29);
    m.lngg = 56; m.lngb = 57; m.lntg = 58; m.lntb = 59;
    m.wig = 60;  m.big = 61;  m.wit = 62;  m.bit = 63;
    m.owg = 64;  m.obg = 65;  m.owt = 66;  m.obt = 67;
    m.egts = 68; m.egtd = 69; m.etgs = 70; m.etgd = 71; m.eggs = 72; m.eggd = 73;
  }
  return m;
}

static inline void gemm_bias(hipStream_t st, const float* A, const float* W,
                             const float* b, float* C, int M, int K) {
  dim3 blk(32, 4), grd(M / 16, 2);
  hipLaunchKernelGGL((wmma_gemm<0>), grd, blk, 0, st, A, W, b, C, M, K, HID,
                     (const float*)nullptr, (const float*)nullptr);
}
static inline void gemm_skip(hipStream_t st, const float* A, const float* W,
                             const float* b, float* C, int M,
                             const float* hprev, const float* skip) {
  dim3 blk(32, 4), grd(M / 16, 2);
  hipLaunchKernelGGL((wmma_gemm<1>), grd, blk, 0, st, A, W, b, C, M, HID, HID,
                     hprev, skip);
}

extern "C" void kernel_launch(void* const* d_in, const int* in_sizes, int n_in,
                              void* d_out, int out_size, void* d_ws,
                              size_t ws_size, hipStream_t stream) {
  const Map mp = resolveMap(in_sizes);
  auto F = [&](int i) { return (const float*)d_in[i]; };
  auto I = [&](int i) { return (const int*)d_in[i]; };
  const int E = in_sizes[mp.egts];  // 500000

  // ---- workspace carve (floats) ----
  float* p = (float*)d_ws;
  auto take = [&](size_t n) { float* r = p; p += n; return r; };
  float* h_g = take((size_t)NGENE * HID);
  float* h_t = take((size_t)NTRAIT * HID);
  float* q_g = take((size_t)NGENE * HID);
  float* q_t = take((size_t)NTRAIT * HID);
  float* k_g = take((size_t)NGENE * HID);
  float* k_t = take((size_t)NTRAIT * HID);
  float* v_g = take((size_t)NGENE * HID);
  float* v_t = take((size_t)NTRAIT * HID);
  float* krel = take((size_t)NGENE * HID);      // reused per edge type
  float* vrel_gt = take((size_t)NGENE * HID);   // src = gene
  float* vrel_tg = take((size_t)NTRAIT * HID);  // src = trait
  float* vrel_gg = take((size_t)NGENE * HID);   // src = gene
  float* s_gt = take((size_t)E * 4);
  float* s_tg = take((size_t)E * 4);
  float* s_gg = take((size_t)E * 4);
  unsigned* m_g = (unsigned*)take((size_t)NGENE * 4);
  unsigned* m_t = (unsigned*)take((size_t)NTRAIT * 4);
  float* den_g = take((size_t)NGENE * 4);
  float* den_t = take((size_t)NTRAIT * 4);
  float* agg_g = take((size_t)NGENE * HID);
  float* agg_t = take((size_t)NTRAIT * HID);

  const dim3 lnb(32, 8);
  const int eBlocks = (E * 4 + 255) / 256;        // score/expsum
  const int aBlocks = (E * 4 + 7) / 8;            // wave-per-(e,h) agg

  // ---- input projections + layernorm ----
  gemm_bias(stream, F(mp.xg), F(mp.wig), F(mp.big), h_g, NGENE, 512);
  gemm_bias(stream, F(mp.xt), F(mp.wit), F(mp.bit), h_t, NTRAIT, 256);
  hipLaunchKernelGGL(layernorm128, dim3(NGENE / 8), lnb, 0, stream, h_g,
                     F(mp.lngg), F(mp.lngb), NGENE);
  hipLaunchKernelGGL(layernorm128, dim3(NTRAIT / 8), lnb, 0, stream, h_t,
                     F(mp.lntg), F(mp.lntb), NTRAIT);

  for (int l = 0; l < 2; ++l) {
    const ConvIdx& c = mp.c[l];
    // K/Q/V projections
    gemm_bias(stream, h_g, F(c.kgw), F(c.kgb), k_g, NGENE, HID);
    gemm_bias(stream, h_g, F(c.qgw), F(c.qgb), q_g, NGENE, HID);
    gemm_bias(stream, h_g, F(c.vgw), F(c.vgb), v_g, NGENE, HID);
    gemm_bias(stream, h_t, F(c.ktw), F(c.ktb), k_t, NTRAIT, HID);
    gemm_bias(stream, h_t, F(c.qtw), F(c.qtb), q_t, NTRAIT, HID);
    gemm_bias(stream, h_t, F(c.vtw), F(c.vtb), v_t, NTRAIT, HID);
    // reset segment-softmax accumulators (fresh every call; graph-safe memsets)
    hipMemsetAsync(m_g, 0, (size_t)NGENE * 4 * 4, stream);
    hipMemsetAsync(m_t, 0, (size_t)NTRAIT * 4 * 4, stream);
    hipMemsetAsync(den_g, 0, (size_t)NGENE * 4 * 4, stream);
    hipMemsetAsync(den_t, 0, (size_t)NTRAIT * 4 * 4, stream);
    hipMemsetAsync(agg_g, 0, (size_t)NGENE * HID * 4, stream);
    hipMemsetAsync(agg_t, 0, (size_t)NTRAIT * HID * 4, stream);

    // gt: gene -> trait
    hipLaunchKernelGGL(wmma_rel, dim3(NGENE / 16), dim3(32, 8), 0, stream, k_g,
                       F(c.agt), krel, NGENE);
    hipLaunchKernelGGL(score_k, dim3(eBlocks), dim3(256), 0, stream,
                       I(mp.egts), I(mp.egtd), q_t, krel, F(c.pgt), s_gt, m_t, E);
    hipLaunchKernelGGL(wmma_rel, dim3(NGENE / 16), dim3(32, 8), 0, stream, v_g,
                       F(c.mgt), vrel_gt, NGENE);
    // tg: trait -> gene
    hipLaunchKernelGGL(wmma_rel, dim3(NTRAIT / 16), dim3(32, 8), 0, stream, k_t,
                       F(c.atg), krel, NTRAIT);
    hipLaunchKernelGGL(score_k, dim3(eBlocks), dim3(256), 0, stream,
                       I(mp.etgs), I(mp.etgd), q_g, krel, F(c.ptg), s_tg, m_g, E);
    hipLaunchKernelGGL(wmma_rel, dim3(NTRAIT / 16), dim3(32, 8), 0, stream, v_t,
                       F(c.mtg), vrel_tg, NTRAIT);
    // gg: gene -> gene
    hipLaunchKernelGGL(wmma_rel, dim3(NGENE / 16), dim3(32, 8), 0, stream, k_g,
                       F(c.agg_), krel, NGENE);
    hipLaunchKernelGGL(score_k, dim3(eBlocks), dim3(256), 0, stream,
                       I(mp.eggs), I(mp.eggd), q_g, krel, F(c.pgg), s_gg, m_g, E);
    hipLaunchKernelGGL(wmma_rel, dim3(NGENE / 16), dim3(32, 8), 0, stream, v_g,
                       F(c.mgg), vrel_gg, NGENE);

    // exp & denominator (softmax is over ALL incoming edge types per dst)
    hipLaunchKernelGGL(expsum_k, dim3(eBlocks), dim3(256), 0, stream,
                       I(mp.egtd), s_gt, m_t, den_t, E);
    hipLaunchKernelGGL(expsum_k, dim3(eBlocks), dim3(256), 0, stream,
                       I(mp.etgd), s_tg, m_g, den_g, E);
    hipLaunchKernelGGL(expsum_k, dim3(eBlocks), dim3(256), 0, stream,
                       I(mp.eggd), s_gg, m_g, den_g, E);
    // weighted aggregation
    hipLaunchKernelGGL(agg_k, dim3(aBlocks), dim3(256), 0, stream, I(mp.egts),
                       I(mp.egtd), s_gt, den_t, vrel_gt, agg_t, E);
    hipLaunchKernelGGL(agg_k, dim3(aBlocks), dim3(256), 0, stream, I(mp.etgs),
                       I(mp.etgd), s_tg, den_g, vrel_tg, agg_g, E);
    hipLaunchKernelGGL(agg_k, dim3(aBlocks), dim3(256), 0, stream, I(mp.eggs),
                       I(mp.eggd), s_gg, den_g, vrel_gg, agg_g, E);

    // gelu -> out-proj with fused skip-blend + residual + relu (in place on h)
    hipLaunchKernelGGL(gelu_k, dim3((NGENE * HID + 255) / 256), dim3(256), 0,
                       stream, agg_g, NGENE * HID);
    hipLaunchKernelGGL(gelu_k, dim3((NTRAIT * HID + 255) / 256), dim3(256), 0,
                       stream, agg_t, NTRAIT * HID);
    gemm_skip(stream, agg_g, F(c.ogw), F(c.ogb), h_g, NGENE, h_g, F(c.skg));
    gemm_skip(stream, agg_t, F(c.otw), F(c.otb), h_t, NTRAIT, h_t, F(c.skt));
  }

  // ---- final output projections, concatenated [gene | trait] ----
  float* out = (float*)d_out;
  gemm_bias(stream, h_g, F(mp.owg), F(mp.obg), out, NGENE, HID);
  gemm_bias(stream, h_t, F(mp.owt), F(mp.obt), out + (size_t)NGENE * HID,
            NTRAIT, HID);
}